// MatchNet_1563368096436
// MI455X (gfx1250) — compile-verified
//
#include <hip/hip_runtime.h>
#include <math.h>

typedef float v2f __attribute__((ext_vector_type(2)));
typedef float v8f __attribute__((ext_vector_type(8)));

#define NPTS 8192
#define NCHUNK 4                               // column chunks per row-tile
#define TILES_PER_CHUNK (NPTS / 16 / NCHUNK)   // 128 16-wide column tiles per chunk
#define WAVES_PER_BLOCK 8
#define EPSV 1e-5f
#define FACTV 2.0f
#define LOG2E 1.4426950408889634f

// ws layout: [0, 8192*4*6 floats) partial softmax states,
//            then float4[8192] packed pc2 (AoS).   Total 896 KB.
#define WS_PACK_OFFSET (NPTS * NCHUNK * 6)

// rcp + one Newton-Raphson step: ~1 ulp f32 reciprocal (matches 1/x accuracy)
__device__ __forceinline__ float fast_rcp(float x) {
    float r = __builtin_amdgcn_rcpf(x);
    return fmaf(fmaf(-x, r, 1.0f), r, r);
}

// Transpose pc2 (4, 8192) SoA -> AoS float4[8192] so the hot loop does one
// global_load_b128 per lane per tile instead of four strided b32 loads.
__global__ __launch_bounds__(256) void matchnet_pack(
    const float* __restrict__ pc2, float4* __restrict__ pack)
{
    const int m = blockIdx.x * blockDim.x + threadIdx.x;
    if (m >= NPTS) return;
    float4 v;
    v.x = pc2[0 * NPTS + m];
    v.y = pc2[1 * NPTS + m];
    v.z = pc2[2 * NPTS + m];
    v.w = pc2[3 * NPTS + m];
    pack[m] = v;
}

// Fused distance + two-pass softmax + weighted-accumulate kernel.
// One wave32 handles one (16-row tile, column chunk) task.
// Pass 1: row-min of d_sq (no trans ops) -> row max of s = FACT/clip(d_sq)
// Pass 2: e = exp2(FACT*log2e*rcp - smax*log2e); accumulate sum + 4 channels.
// Partial state per row per chunk: {smax, sum, acc0..acc3} -> 6 floats in ws.
__global__ __launch_bounds__(256) void matchnet_fused(
    const float* __restrict__ pc1, const float4* __restrict__ p2,
    float* __restrict__ ws)
{
    const int tid  = threadIdx.x;
    const int wave = tid >> 5;
    const int lane = tid & 31;
    const int l    = lane & 15;     // column-within-tile / row-within-tile index
    const int half = lane >> 4;     // 0: K=0,1 & rows 0-7 ; 1: K=2,3 & rows 8-15

    const int task    = blockIdx.x * WAVES_PER_BLOCK + wave;  // 0..2047
    const int rowTile = task >> 2;                            // 0..511
    const int chunk   = task & (NCHUNK - 1);                  // 0..3
    const int n0      = rowTile * 16;

    // ---- A matrix (16x4 f32), row M = l.  A[n,:] = [-2x, -2y, -2z, sq1[n]] ----
    const float x1 = pc1[0 * NPTS + n0 + l];
    const float y1 = pc1[1 * NPTS + n0 + l];
    const float z1 = pc1[2 * NPTS + n0 + l];
    const float sq1 = x1 * x1 + y1 * y1 + z1 * z1;
    v2f a;
    a.x = half ? (-2.0f * z1) : (-2.0f * x1);   // K=2 : K=0
    a.y = half ? sq1          : (-2.0f * y1);   // K=3 : K=1

    const int mbase = chunk * (TILES_PER_CHUNK * 16);

    // ================= PASS 1: per-row min of d_sq =================
    float mind[8];
#pragma unroll
    for (int r = 0; r < 8; ++r) mind[r] = INFINITY;

    for (int t = 0; t < TILES_PER_CHUNK; ++t) {
        const int m = mbase + t * 16 + l;
        const float4 q = p2[m];                 // one b128 load

        v2f b;
        b.x = half ? q.z  : q.x;                // K=2 : K=0
        b.y = half ? 1.0f : q.y;                // K=3 : K=1

        v8f c = {0.0f, 0.0f, 0.0f, 0.0f, 0.0f, 0.0f, 0.0f, 0.0f};
        v8f d = __builtin_amdgcn_wmma_f32_16x16x4_f32(
            false, a, false, b, (short)0, c, false, false);

        const float sq2 = q.x * q.x + q.y * q.y + q.z * q.z;
#pragma unroll
        for (int r = 0; r < 8; ++r)
            mind[r] = fminf(mind[r], d[r] + sq2);
    }

    // min-butterfly across the 16 lanes of each half-wave
#pragma unroll
    for (int mask = 1; mask < 16; mask <<= 1) {
#pragma unroll
        for (int r = 0; r < 8; ++r)
            mind[r] = fminf(mind[r], __shfl_xor(mind[r], mask, 32));
    }

    // row max of s = FACT / clip(min d_sq, EPS); also base-2 scaled copy
    float smax[8], smax2[8];
#pragma unroll
    for (int r = 0; r < 8; ++r) {
        smax[r]  = FACTV * fast_rcp(fmaxf(mind[r], EPSV));
        smax2[r] = smax[r] * LOG2E;
    }

    // ================= PASS 2: exp + weighted accumulate =================
    float sumr[8], acc[8][4];
#pragma unroll
    for (int r = 0; r < 8; ++r) {
        sumr[r] = 0.0f;
        acc[r][0] = acc[r][1] = acc[r][2] = acc[r][3] = 0.0f;
    }

    const float K2 = FACTV * LOG2E;
    for (int t = 0; t < TILES_PER_CHUNK; ++t) {
        const int m = mbase + t * 16 + l;
        const float4 q = p2[m];                 // one b128 load

        v2f b;
        b.x = half ? q.z  : q.x;
        b.y = half ? 1.0f : q.y;

        v8f c = {0.0f, 0.0f, 0.0f, 0.0f, 0.0f, 0.0f, 0.0f, 0.0f};
        v8f d = __builtin_amdgcn_wmma_f32_16x16x4_f32(
            false, a, false, b, (short)0, c, false, false);

        const float sq2 = q.x * q.x + q.y * q.y + q.z * q.z;
#pragma unroll
        for (int r = 0; r < 8; ++r) {
            const float dsq = fmaxf(d[r] + sq2, EPSV);
            const float rc  = fast_rcp(dsq);
            // e = exp(FACT*rc - smax) = exp2(K2*rc - smax2)   (<= 1, no overflow)
            const float e   = __builtin_amdgcn_exp2f(fmaf(K2, rc, -smax2[r]));
            sumr[r]   += e;
            acc[r][0] += e * q.x;
            acc[r][1] += e * q.y;
            acc[r][2] += e * q.z;
            acc[r][3] += e * q.w;
        }
    }

    // plain sum-butterfly across the 16 lanes (same smax everywhere -> just add)
#pragma unroll
    for (int mask = 1; mask < 16; mask <<= 1) {
#pragma unroll
        for (int r = 0; r < 8; ++r) {
            sumr[r]   += __shfl_xor(sumr[r],   mask, 32);
            acc[r][0] += __shfl_xor(acc[r][0], mask, 32);
            acc[r][1] += __shfl_xor(acc[r][1], mask, 32);
            acc[r][2] += __shfl_xor(acc[r][2], mask, 32);
            acc[r][3] += __shfl_xor(acc[r][3], mask, 32);
        }
    }

    // ---- one lane per half writes the 8 row partials for this chunk ----
    if (l == 0) {
#pragma unroll
        for (int r = 0; r < 8; ++r) {
            const int n = n0 + r + 8 * half;
            float* p = ws + ((size_t)n * NCHUNK + chunk) * 6;
            p[0] = smax[r];
            p[1] = sumr[r];
            p[2] = acc[r][0];
            p[3] = acc[r][1];
            p[4] = acc[r][2];
            p[5] = acc[r][3];
        }
    }
}

// Merge the NCHUNK partial softmax states per row (LSE combine), normalize, write.
__global__ __launch_bounds__(256) void matchnet_merge(
    const float* __restrict__ ws, float* __restrict__ out)
{
    const int n = blockIdx.x * blockDim.x + threadIdx.x;
    if (n >= NPTS) return;

    float m = -INFINITY, s = 0.0f, a0 = 0.0f, a1 = 0.0f, a2 = 0.0f, a3 = 0.0f;
#pragma unroll
    for (int c = 0; c < NCHUNK; ++c) {
        const float* p = ws + ((size_t)n * NCHUNK + c) * 6;
        const float pm = p[0];
        const float nm = fmaxf(m, pm);
        const float f1 = __expf(m - nm);
        const float f2 = __expf(pm - nm);
        s  = s  * f1 + p[1] * f2;
        a0 = a0 * f1 + p[2] * f2;
        a1 = a1 * f1 + p[3] * f2;
        a2 = a2 * f1 + p[4] * f2;
        a3 = a3 * f1 + p[5] * f2;
        m = nm;
    }
    const float inv = 1.0f / s;
    out[0 * NPTS + n] = a0 * inv;
    out[1 * NPTS + n] = a1 * inv;
    out[2 * NPTS + n] = a2 * inv;
    out[3 * NPTS + n] = a3 * inv;
    out[4 * NPTS + n] = 1.0f;          // indexor
}

extern "C" void kernel_launch(void* const* d_in, const int* in_sizes, int n_in,
                              void* d_out, int out_size, void* d_ws, size_t ws_size,
                              hipStream_t stream) {
    (void)in_sizes; (void)n_in; (void)out_size; (void)ws_size;
    const float* pc1 = (const float*)d_in[0];
    const float* pc2 = (const float*)d_in[1];
    float* out  = (float*)d_out;
    float* ws   = (float*)d_ws;                      // needs 896 KB
    float4* pack = (float4*)(ws + WS_PACK_OFFSET);   // 16B-aligned (768KB offset)

    hipLaunchKernelGGL(matchnet_pack, dim3((NPTS + 255) / 256), dim3(256), 0, stream,
                       pc2, pack);

    const int tasks  = (NPTS / 16) * NCHUNK;          // 2048 wave-tasks
    const int blocks = tasks / WAVES_PER_BLOCK;       // 256 blocks of 8 waves
    hipLaunchKernelGGL(matchnet_fused, dim3(blocks), dim3(256), 0, stream,
                       pc1, pack, ws);

    hipLaunchKernelGGL(matchnet_merge, dim3((NPTS + 255) / 256), dim3(256), 0, stream,
                       ws, out);
}